// ACNetworkBase_85203561218284
// MI455X (gfx1250) — compile-verified
//
#include <hip/hip_runtime.h>
#include <cmath>

// ---------------------------------------------------------------------------
// Discounted-returns scan + normalize for MI455X (gfx1250).
// Bandwidth-bound (0.25 FLOP/byte << roofline knee at 23.3 TB/s):
//   * TDM tensor_load_to_lds (async DMA, TENSORcnt) stages 2 KiB/wave tiles,
//     with TDM LDS padding (1 dword per 16) for bank-conflict-free reads.
//   * b128 global accesses everywhere; NT stores for the final output.
//   * 2 sweeps only (in+local-write, L2-resident re-read+normalize): the
//     global moments are closed ANALYTICALLY from per-chunk reductions
//     (sum r = A + C*G1, sum r^2 = Q + 2*C*B + C^2*G2), so no moment sweep.
// Assumes T % 4096 == 0 (reference fixes T = 2^25).
// ---------------------------------------------------------------------------

#define GAMMA 0.99f
#define LN_GAMMA (-0.01005033585350145f)
#define EPSN 0.0001f

constexpr int BLOCK = 256;     // 8 waves (wave32)
constexpr int PER_THREAD = 16;
constexpr int CHUNK = BLOCK * PER_THREAD;          // 4096 elements / block
constexpr int WAVE_ELEMS = 32 * PER_THREAD;        // 512 elements / wave
constexpr int WAVE_PAD_ELEMS = WAVE_ELEMS + WAVE_ELEMS / 16;  // 544 (TDM padded)

typedef unsigned int u32x4 __attribute__((ext_vector_type(4)));
typedef int i32x4 __attribute__((ext_vector_type(4)));
typedef int i32x8 __attribute__((ext_vector_type(8)));
typedef float f32x4 __attribute__((ext_vector_type(4)));

// ---------------------------------------------------------------------------
// Pass 1: chunk-local discounted suffix scan + per-chunk moment components.
//   out[j]   = local_j = sum_{k>=j, same chunk} gamma^(k-j) * r[k]
//   chunkS[b] = local_0            (full chunk discounted sum)
//   chunkA[b] = sum_j local_j
//   chunkB[b] = sum_j local_j * gamma^(L-j)
//   chunkQ[b] = sum_j local_j^2
// ---------------------------------------------------------------------------
__global__ void k_local_scan(const float* __restrict__ rw,
                             float* __restrict__ out,
                             float* __restrict__ chunkS,
                             float* __restrict__ chunkA,
                             float* __restrict__ chunkB,
                             float* __restrict__ chunkQ)
{
    __shared__ float tile[8 * WAVE_PAD_ELEMS];   // 17408 B, TDM-padded layout
    __shared__ float ssum[BLOCK];
    __shared__ float ra[BLOCK];
    __shared__ float rb[BLOCK];
    __shared__ float rq[BLOCK];

    const int b = blockIdx.x;
    const int t = threadIdx.x;
    const long long base = (long long)b * CHUNK;

    // ---- TDM: each wave DMAs its 512-float slice global -> LDS ------------
    {
        const int wave = __builtin_amdgcn_readfirstlane(t >> 5);
        unsigned ldsOff = (unsigned)(size_t)(&tile[0])
                        + (unsigned)(wave * WAVE_PAD_ELEMS * 4);
        unsigned long long ga =
            (unsigned long long)(size_t)(rw + base + (long long)wave * WAVE_ELEMS);

        // D# group 0: count=1 | lds_addr | global_addr | type=2
        u32x4 g0 = { 1u,
                     ldsOff,
                     (unsigned)(ga & 0xffffffffull),
                     ((unsigned)(ga >> 32) & 0x01ffffffu) | 0x80000000u };
        // D# group 1: data_size=4B (2<<16), pad_enable (1<<20),
        //             pad_interval=16 dwords (3<<22), pad_amount=1 dword (0<<25)
        i32x8 g1 = { 0x00D20000,
                     (WAVE_ELEMS << 16),        // tensor_dim0 lo16
                     (1 << 16),                 // tensor_dim0 hi=0 | tensor_dim1=1
                     (WAVE_ELEMS << 16),        // tensor_dim1 hi=0 | tile_dim0=512
                     1,                         // tile_dim1=1, tile_dim2=0
                     WAVE_ELEMS,                // tensor_dim0_stride lo32
                     0, 0 };
        i32x4 z4 = { 0, 0, 0, 0 };
#if __clang_major__ >= 23
        i32x8 z8 = { 0, 0, 0, 0, 0, 0, 0, 0 };
        __builtin_amdgcn_tensor_load_to_lds(g0, g1, z4, z4, z8, 0);
#else
        __builtin_amdgcn_tensor_load_to_lds(g0, g1, z4, z4, 0);
#endif
        __builtin_amdgcn_s_wait_tensorcnt(0);
    }
    __syncthreads();

    // ---- per-thread serial suffix scan over 16 contiguous elements --------
    // Padded LDS layout: lane element j -> word (t>>5)*544 + (t&31)*17 + j.
    // Stride 17 (odd) => 32 lanes hit 32 distinct banks: conflict-free.
    const float* src = &tile[(t >> 5) * WAVE_PAD_ELEMS + (t & 31) * 17];
    float v[PER_THREAD];
    float acc = 0.f;
#pragma unroll
    for (int i = PER_THREAD - 1; i >= 0; --i) {
        acc = src[i] + GAMMA * acc;
        v[i] = acc;
    }
    ssum[t] = acc;          // discounted sum of this thread's 16 elements
    __syncthreads();

    // ---- cross-thread suffix scan, ratio gamma^16 (Hillis-Steele) ---------
    const float g2 = GAMMA * GAMMA, g4 = g2 * g2, g8 = g4 * g4, g16 = g8 * g8;
    float qq = g16;
    float y = acc;
    for (int off = 1; off < BLOCK; off <<= 1) {
        float other = (t + off < BLOCK) ? ssum[t + off] : 0.f;
        y += qq * other;
        __syncthreads();
        ssum[t] = y;
        __syncthreads();
        qq *= qq;
    }
    const float carry = (t < BLOCK - 1) ? ssum[t + 1] : 0.f;

    // ---- fold carry in: element i gets gamma^(16-i) * carry ---------------
    float w = GAMMA;
#pragma unroll
    for (int i = PER_THREAD - 1; i >= 0; --i) {
        v[i] += w * carry;
        w *= GAMMA;
    }

    // ---- per-thread moment partials: A = sum v, B = sum v*g^(L-j), Q = sum v^2
    float wB = __expf((float)(CHUNK - t * PER_THREAD - 15) * LN_GAMMA);
    float pa = 0.f, pb = 0.f, pq = 0.f;
#pragma unroll
    for (int i = PER_THREAD - 1; i >= 0; --i) {
        pa += v[i];
        pb += v[i] * wB;
        pq += v[i] * v[i];
        wB *= GAMMA;
    }
    ra[t] = pa; rb[t] = pb; rq[t] = pq;

    // ---- b128 stores of chunk-local returns (temporal: re-read once, and
    //      the 128 MB intermediate fits the 192 MB L2) ----------------------
    f32x4* o = (f32x4*)(out + base + t * PER_THREAD);
    f32x4 a0 = { v[0],  v[1],  v[2],  v[3]  };
    f32x4 a1 = { v[4],  v[5],  v[6],  v[7]  };
    f32x4 a2 = { v[8],  v[9],  v[10], v[11] };
    f32x4 a3 = { v[12], v[13], v[14], v[15] };
    o[0] = a0; o[1] = a1; o[2] = a2; o[3] = a3;

    // ---- tree-reduce the three moment components --------------------------
    __syncthreads();
    for (int off = BLOCK / 2; off > 0; off >>= 1) {
        if (t < off) {
            ra[t] += ra[t + off];
            rb[t] += rb[t + off];
            rq[t] += rq[t + off];
        }
        __syncthreads();
    }
    if (t == 0) {
        chunkS[b] = ssum[0];
        chunkA[b] = ra[0];
        chunkB[b] = rb[0];
        chunkQ[b] = rq[0];
    }
}

// ---------------------------------------------------------------------------
// Pass 2 (1 block): suffix scan over chunk sums with ratio Qg = gamma^4096
// producing per-chunk carries C[b] = S[b+1] + Qg*C[b+1], AND closes the
// global moments analytically in f64:
//   sum   += A[b] + C[b]*G1
//   sumsq += Q[b] + 2*C[b]*B[b] + C[b]^2*G2
// Writes stats = { mean, 1/(std+eps) }.
// ---------------------------------------------------------------------------
__global__ void k_chunk_scan(const float* __restrict__ S,
                             const float* __restrict__ A,
                             const float* __restrict__ Bm,
                             const float* __restrict__ Qm,
                             float* __restrict__ C,
                             float* __restrict__ stats,
                             int nchunks, float Qg,
                             double G1, double G2, double invN)
{
    __shared__ float s2[BLOCK];
    __shared__ double rs[BLOCK];
    __shared__ double rq2[BLOCK];
    const int t = threadIdx.x;
    constexpr int PER = 32;
    const int base = t * PER;

    float vals[PER];
#pragma unroll
    for (int i = 0; i < PER; ++i) {
        int idx = base + i;
        vals[i] = (idx < nchunks) ? S[idx] : 0.f;
    }
    float acc = 0.f;
#pragma unroll
    for (int i = PER - 1; i >= 0; --i) acc = vals[i] + Qg * acc;
    s2[t] = acc;
    __syncthreads();

    float Qp = Qg;                      // Qg^32 (underflows to 0: exact enough)
#pragma unroll
    for (int k = 0; k < 5; ++k) Qp *= Qp;
    float y = acc;
    for (int off = 1; off < BLOCK; off <<= 1) {
        float other = (t + off < BLOCK) ? s2[t + off] : 0.f;
        y += Qp * other;
        __syncthreads();
        s2[t] = y;
        __syncthreads();
        Qp *= Qp;
    }
    float c = (t < BLOCK - 1) ? s2[t + 1] : 0.f;

    double cs = 0.0, cq = 0.0;
#pragma unroll
    for (int i = PER - 1; i >= 0; --i) {
        int idx = base + i;
        if (idx < nchunks) {
            C[idx] = c;
            double cd = (double)c;
            cs += (double)A[idx] + cd * G1;
            cq += (double)Qm[idx] + 2.0 * cd * (double)Bm[idx] + cd * cd * G2;
        }
        c = vals[i] + Qg * c;
    }
    rs[t] = cs; rq2[t] = cq;
    __syncthreads();
    for (int off = BLOCK / 2; off > 0; off >>= 1) {
        if (t < off) { rs[t] += rs[t + off]; rq2[t] += rq2[t + off]; }
        __syncthreads();
    }
    if (t == 0) {
        double mean = rs[0] * invN;
        double var  = rq2[0] * invN - mean * mean;
        if (var < 0.0) var = 0.0;
        stats[0] = (float)mean;
        stats[1] = 1.0f / ((float)sqrt(var) + EPSN);
    }
}

// ---------------------------------------------------------------------------
// Pass 3: apply carry + normalization in-place; NT b128 stores (final data,
// never re-read by us -> keep L2 for the intermediate working set).
// ---------------------------------------------------------------------------
__global__ void k_norm(float* __restrict__ data,
                       const float* __restrict__ C,
                       const float* __restrict__ stats)
{
    const int b = blockIdx.x, t = threadIdx.x;
    const long long base = (long long)b * CHUNK;
    const float carry = C[b];
    const float mean = stats[0];
    const float inv  = stats[1];

    float w = __expf((float)(CHUNK - t * PER_THREAD - 15) * LN_GAMMA);
    f32x4* p = (f32x4*)(data + base + t * PER_THREAD);
    f32x4 x0 = p[0], x1 = p[1], x2 = p[2], x3 = p[3];
    float v[PER_THREAD] = { x0.x, x0.y, x0.z, x0.w,  x1.x, x1.y, x1.z, x1.w,
                            x2.x, x2.y, x2.z, x2.w,  x3.x, x3.y, x3.z, x3.w };
#pragma unroll
    for (int i = PER_THREAD - 1; i >= 0; --i) {
        v[i] = (v[i] + w * carry - mean) * inv;
        w *= GAMMA;
    }
    f32x4 a0 = { v[0],  v[1],  v[2],  v[3]  };
    f32x4 a1 = { v[4],  v[5],  v[6],  v[7]  };
    f32x4 a2 = { v[8],  v[9],  v[10], v[11] };
    f32x4 a3 = { v[12], v[13], v[14], v[15] };
    __builtin_nontemporal_store(a0, p + 0);
    __builtin_nontemporal_store(a1, p + 1);
    __builtin_nontemporal_store(a2, p + 2);
    __builtin_nontemporal_store(a3, p + 3);
}

// ---------------------------------------------------------------------------
extern "C" void kernel_launch(void* const* d_in, const int* in_sizes, int n_in,
                              void* d_out, int out_size, void* d_ws, size_t ws_size,
                              hipStream_t stream)
{
    const float* rewards = (const float*)d_in[0];
    float* out = (float*)d_out;
    const int n = in_sizes[0];
    const int nchunks = n / CHUNK;               // T = 2^25 -> 8192

    char* ws = (char*)d_ws;
    float* stats  = (float*)ws;                  // mean, 1/(std+eps)
    float* chunkS = (float*)(ws + 64);
    float* chunkC = chunkS + nchunks;
    float* chunkA = chunkC + nchunks;
    float* chunkB = chunkA + nchunks;
    float* chunkQ = chunkB + nchunks;

    const double g  = (double)GAMMA;
    const double gL = pow(g, (double)CHUNK);                     // gamma^4096
    const float  Qg = (float)gL;
    const double G1 = g * (1.0 - gL) / (1.0 - g);                // sum g^k, k=1..L
    const double G2 = g * g * (1.0 - gL * gL) / (1.0 - g * g);   // sum g^2k
    const double invN = 1.0 / (double)n;

    k_local_scan<<<nchunks, BLOCK, 0, stream>>>(rewards, out, chunkS,
                                                chunkA, chunkB, chunkQ);
    k_chunk_scan<<<1, BLOCK, 0, stream>>>(chunkS, chunkA, chunkB, chunkQ,
                                          chunkC, stats, nchunks, Qg,
                                          G1, G2, invN);
    k_norm<<<nchunks, BLOCK, 0, stream>>>(out, chunkC, stats);
}